// cubic_attention_68796786147852
// MI455X (gfx1250) — compile-verified
//
#include <hip/hip_runtime.h>
#include <math.h>

typedef float v2f __attribute__((ext_vector_type(2)));
typedef float v8f __attribute__((ext_vector_type(8)));

#define N_    4
#define C_    64
#define H_    256
#define W_    256
#define HW_   65536
#define CHW_  (C_ * HW_)
#define NH_   10          // NUM_HASHES == GROUP*K == 10
#define K_    5
#define NBINS 1024
#define NSB   256         // sort blocks (256 pixels each)
#define ALDS  65          // padded LDS stride for A-matrix (bank-conflict free)

// ---------- 1) per-batch sum of squares (partials, fixed-order -> deterministic)
__global__ void k_sumsq(const float* __restrict__ x, float* __restrict__ part) {
    int b = blockIdx.y, chunk = blockIdx.x, tid = threadIdx.x;
    const float* p = x + (size_t)b * CHW_ + (size_t)chunk * 16384;
    float s = 0.f;
    for (int i = tid; i < 16384; i += 256) { float v = p[i]; s += v * v; }
    __shared__ float sh[256];
    sh[tid] = s; __syncthreads();
    for (int off = 128; off > 0; off >>= 1) {
        if (tid < off) sh[tid] += sh[tid + off];
        __syncthreads();
    }
    if (tid == 0) part[b * 256 + chunk] = sh[0];
}

// ---------- 2) reduce partials, argmax over batches (ties -> lowest index, like jnp.argmax)
__global__ void k_best(const float* __restrict__ part, int* __restrict__ best) {
    __shared__ float sh[256];
    __shared__ float tot[N_];
    int tid = threadIdx.x;
    for (int b = 0; b < N_; ++b) {
        sh[tid] = part[b * 256 + tid]; __syncthreads();
        for (int off = 128; off > 0; off >>= 1) {
            if (tid < off) sh[tid] += sh[tid + off];
            __syncthreads();
        }
        if (tid == 0) tot[b] = sh[0];
        __syncthreads();
    }
    if (tid == 0) {
        int bi = 0; float bv = tot[0];
        for (int b = 1; b < N_; ++b) if (tot[b] > bv) { bv = tot[b]; bi = b; }
        *best = bi;
    }
}

// ---------- 3/7) WMMA GEMM: D[16x16] = A(10x64 padded) x B(64 x 16-col tile of best batch)
// GATHER=false: A=rvecs, epilogue = LSH bucket keys.  GATHER=true: A=conv_w, B cols
// permuted by `order`, epilogue = sigmoid -> filt[10][hw].
template <bool GATHER>
__global__ void k_gemm10(const float* __restrict__ X, const float* __restrict__ Wrow,
                         const int* __restrict__ bestp, const unsigned* __restrict__ order,
                         unsigned* __restrict__ keys, float* __restrict__ filt) {
    __shared__ float As[16 * ALDS];
    int tid = threadIdx.x;
    for (int i = tid; i < 16 * ALDS; i += 256) {
        int r = i / ALDS, k = i - r * ALDS;
        As[i] = (r < NH_ && k < 64) ? Wrow[r * 64 + k] : 0.f;
    }
    __syncthreads();

    int wave = tid >> 5, lane = tid & 31;
    int tile = blockIdx.x * 8 + wave;           // 4096 tiles of 16 columns
    bool low = lane < 16;
    int  l   = lane & 15;
    int  j   = tile * 16 + l;                   // output column (pixel)
    const float* Xb = X + (size_t)(*bestp) * CHW_;
    int src = GATHER ? (int)order[j] : j;       // gathered source pixel

    v8f d = {0.f, 0.f, 0.f, 0.f, 0.f, 0.f, 0.f, 0.f};
#pragma unroll
    for (int kk = 0; kk < 16; ++kk) {
        int ka = 4 * kk + (low ? 0 : 2);
        v2f a; a.x = As[l * ALDS + ka];            a.y = As[l * ALDS + ka + 1];
        v2f b; b.x = Xb[(size_t)ka * HW_ + src];   b.y = Xb[(size_t)(ka + 1) * HW_ + src];
        d = __builtin_amdgcn_wmma_f32_16x16x4_f32(false, a, false, b, (short)0, d,
                                                  false, false);
    }

    if (GATHER) {
        if (low) {
#pragma unroll
            for (int v = 0; v < 8; ++v)
                filt[(size_t)v * HW_ + j] = 1.f / (1.f + expf(-d[v]));
        } else {
            filt[(size_t)8 * HW_ + j] = 1.f / (1.f + expf(-d[0]));
            filt[(size_t)9 * HW_ + j] = 1.f / (1.f + expf(-d[1]));
        }
    } else {
        unsigned bits = 0;
        if (low) {
#pragma unroll
            for (int v = 0; v < 8; ++v) bits |= (d[v] > 0.f ? 1u : 0u) << v;
        } else {
            bits |= (d[0] > 0.f ? 1u : 0u) << 8;
            bits |= (d[1] > 0.f ? 1u : 0u) << 9;
        }
        bits |= __shfl_xor(bits, 16, 32);       // combine rows 0-7 with rows 8-9
        if (low) keys[j] = bits;
    }
}

// ---------- 4) per-block histogram (256 pixels/block, 1024 bins)
__global__ void k_hist(const unsigned* __restrict__ keys, unsigned* __restrict__ hist) {
    __shared__ unsigned h[NBINS];
    int tid = threadIdx.x, blk = blockIdx.x;
    for (int i = tid; i < NBINS; i += 256) h[i] = 0;
    __syncthreads();
    atomicAdd(&h[keys[blk * 256 + tid]], 1u);
    __syncthreads();
    for (int i = tid; i < NBINS; i += 256) hist[i * NSB + blk] = h[i];
}

// ---------- 5) scan: hist[bin][blk] counts -> global stable offsets (in place)
__global__ void k_scan(unsigned* __restrict__ hist) {
    int bin = threadIdx.x;                      // 1024 threads
    unsigned total = 0;
    for (int b = 0; b < NSB; ++b) total += hist[bin * NSB + b];
    __shared__ unsigned sh[NBINS];
    sh[bin] = total; __syncthreads();
    for (int off = 1; off < NBINS; off <<= 1) { // Hillis-Steele inclusive scan
        unsigned v = sh[bin];
        unsigned add = (bin >= off) ? sh[bin - off] : 0u;
        __syncthreads();
        sh[bin] = v + add;
        __syncthreads();
    }
    unsigned run = sh[bin] - total;             // exclusive base for this bin
    for (int b = 0; b < NSB; ++b) {
        unsigned c = hist[bin * NSB + b];
        hist[bin * NSB + b] = run;
        run += c;
    }
}

// ---------- 6) stable scatter -> order[] (serial within a 256-pixel block = stable)
__global__ void k_scatter(const unsigned* __restrict__ keys, const unsigned* __restrict__ hist,
                          unsigned* __restrict__ order) {
    __shared__ unsigned off[NBINS];
    __shared__ unsigned ks[256];
    int tid = threadIdx.x, blk = blockIdx.x;
    for (int i = tid; i < NBINS; i += 256) off[i] = hist[i * NSB + blk];
    ks[tid] = keys[blk * 256 + tid];
    __syncthreads();
    if (tid == 0) {
        for (int i = 0; i < 256; ++i) {
            unsigned k = ks[i];
            order[off[k]++] = (unsigned)(blk * 256 + i);
        }
    }
}

// ---------- 8) strip convolution (reflect pad), optional gamma/beta fuse
template <bool HORIZ, bool FUSE>
__global__ void k_strip(const float* __restrict__ X, const float* __restrict__ filt,
                        float* __restrict__ out, const float* __restrict__ gamma,
                        const float* __restrict__ beta, const float* __restrict__ x0) {
    int idx = blockIdx.x * 256 + threadIdx.x;   // over n*c*h*w
    int wi = idx & (W_ - 1);
    int hi = (idx >> 8) & (H_ - 1);
    int ch = (idx >> 16) & (C_ - 1);
    int p  = (hi << 8) | wi;
    int g  = ch >> 5;                           // channel group (c/GROUP = 32)
    const float* fb = filt + (size_t)g * K_ * HW_ + p;
    float acc = 0.f;
#pragma unroll
    for (int k = 0; k < K_; ++k) {
        int t = (HORIZ ? wi : hi) + k - 2;
        if (t < 0) t = -t;
        if (t > 255) t = 510 - t;               // 'reflect' (no edge repeat)
        float xv = HORIZ ? X[(idx - wi) + t] : X[(idx - p) + t * W_ + wi];
        acc += xv * fb[(size_t)k * HW_];
    }
    if (FUSE) out[idx] = gamma[ch] * acc + beta[ch] * x0[idx];
    else      out[idx] = acc;
}

extern "C" void kernel_launch(void* const* d_in, const int* in_sizes, int n_in,
                              void* d_out, int out_size, void* d_ws, size_t ws_size,
                              hipStream_t stream) {
    const float* x       = (const float*)d_in[0];
    const float* conv_h  = (const float*)d_in[1];
    const float* conv_w  = (const float*)d_in[2];
    const float* rvecs_h = (const float*)d_in[3];
    const float* rvecs_w = (const float*)d_in[4];
    const float* gamma   = (const float*)d_in[5];
    const float* beta    = (const float*)d_in[6];

    char* ws = (char*)d_ws;
    size_t o = 0;
    float*    out1 = (float*)(ws + o);    o += sizeof(float) * (size_t)N_ * CHW_;
    float*    filt = (float*)(ws + o);    o += sizeof(float) * (size_t)NH_ * HW_;
    unsigned* keys = (unsigned*)(ws + o); o += sizeof(unsigned) * HW_;
    unsigned* ord  = (unsigned*)(ws + o); o += sizeof(unsigned) * HW_;
    unsigned* hist = (unsigned*)(ws + o); o += sizeof(unsigned) * NBINS * NSB;
    float*    part = (float*)(ws + o);    o += sizeof(float) * N_ * 256;
    int*      best = (int*)(ws + o);

    const int strip_blocks = (N_ * CHW_) / 256;

    // ---------------- pass 1: horizontal, input = x, output = out1
    k_sumsq  <<<dim3(256, N_), 256, 0, stream>>>(x, part);
    k_best   <<<1, 256, 0, stream>>>(part, best);
    k_gemm10<false><<<512, 256, 0, stream>>>(x, rvecs_h, best, nullptr, keys, nullptr);
    k_hist   <<<NSB, 256, 0, stream>>>(keys, hist);
    k_scan   <<<1, NBINS, 0, stream>>>(hist);
    k_scatter<<<NSB, 256, 0, stream>>>(keys, hist, ord);
    k_gemm10<true><<<512, 256, 0, stream>>>(x, conv_h, best, ord, nullptr, filt);
    k_strip<true, false><<<strip_blocks, 256, 0, stream>>>(x, filt, out1, gamma, beta, x);

    // ---------------- pass 2: vertical, input = out1, output = d_out (+gamma/beta fuse)
    k_sumsq  <<<dim3(256, N_), 256, 0, stream>>>(out1, part);
    k_best   <<<1, 256, 0, stream>>>(part, best);
    k_gemm10<false><<<512, 256, 0, stream>>>(out1, rvecs_w, best, nullptr, keys, nullptr);
    k_hist   <<<NSB, 256, 0, stream>>>(keys, hist);
    k_scan   <<<1, NBINS, 0, stream>>>(hist);
    k_scatter<<<NSB, 256, 0, stream>>>(keys, hist, ord);
    k_gemm10<true><<<512, 256, 0, stream>>>(out1, conv_w, best, ord, nullptr, filt);
    k_strip<false, true><<<strip_blocks, 256, 0, stream>>>(out1, filt, (float*)d_out,
                                                           gamma, beta, x);
}